// SGSNetLoss_44590350467622
// MI455X (gfx1250) — compile-verified
//
#include <hip/hip_runtime.h>
#include <hip/hip_bf16.h>
#include <stdint.h>

// Problem constants (from reference)
#define BSZ   4096
#define NTGT  32
#define AANC  3
#define HDIM  13
#define WDIM  13
#define CCLS  5
#define CH    (5 + CCLS)                 // 10 channels per anchor
#define HW    (HDIM * WDIM)              // 169
#define CELLS (AANC * HW)                // 507
#define PREDN (AANC * CH * HW)           // 5070 floats per batch item (20280 B)
#define NTH   256

typedef float v2f __attribute__((ext_vector_type(2)));
typedef float v8f __attribute__((ext_vector_type(8)));

#if defined(__has_builtin)
#  if __has_builtin(__builtin_amdgcn_global_load_async_to_lds_b64)
#    define ASYNC_LDS_B64 1
#  elif __has_builtin(__builtin_amdgcn_global_load_async_to_lds_b32)
#    define ASYNC_LDS_B32 1
#  endif
#endif

__device__ __forceinline__ float bce_logits(float x, float t) {
    // max(x,0) - x*t + softplus(-|x|)
    return fmaxf(x, 0.0f) - x * t + log1pf(__expf(-fabsf(x)));
}

// ---------------------------------------------------------------------------
// Kernel 1: one workgroup per batch item.
//   Issue async-to-LDS DMA for the 20KB prediction slice, overlap it with
//   zeroing the LDS target grids, scatter <=32 targets, compute masked
//   BCE/MSE sums, reduce with wave32 shuffles, emit per-batch scalars.
// ---------------------------------------------------------------------------
__global__ __launch_bounds__(NTH) void sgs_loss_main(
    const float* __restrict__ pred,      // (B, A*CH, H, W)
    const float* __restrict__ tboxes,    // (B, N, 4)
    const int*   __restrict__ tlabels,   // (B, N) int32
    const float* __restrict__ anchors,   // (A, 2)
    float* __restrict__ ws_obj,
    float* __restrict__ ws_bbox,
    float* __restrict__ ws_cls)
{
    __shared__ __align__(16) float s_pred[PREDN];   // 5070
    __shared__ float s_tobj[CELLS];                 // 507
    __shared__ float s_tbox[CELLS * 4];             // 2028
    __shared__ float s_tcls[CELLS * CCLS];          // 2535
    __shared__ float s_red[(NTH / 32) * 5];         // 40

    const int b   = blockIdx.x;
    const int tid = threadIdx.x;
    const float* __restrict__ predb = pred + (size_t)b * PREDN;

    // Kick off the prediction-slice DMA into LDS first so it overlaps the
    // LDS zeroing below.  Per-batch slice is 8-byte aligned (20280 % 8 == 0),
    // so b64 granularity is always safe; b128 would not be (20280 % 16 == 8).
#if defined(ASYNC_LDS_B64)
    {
        // Per hipcc diagnostics: param0 is a global pointer to a 2-int
        // GCC-style vector, param1 the LDS-side counterpart.
        typedef int V2I __attribute__((vector_size(8)));
        typedef __attribute__((address_space(1))) V2I GV;
        typedef __attribute__((address_space(3))) V2I LV;
        const int nchunks = PREDN / 2;   // 2535 x 8B
        for (int i = tid; i < nchunks; i += NTH) {
            GV* gp = (GV*)(uintptr_t)(predb + 2 * i);
            LV* lp = (LV*)(uint32_t)(uintptr_t)(&s_pred[2 * i]);
            __builtin_amdgcn_global_load_async_to_lds_b64(gp, lp, 0, 0);
        }
    }
#elif defined(ASYNC_LDS_B32)
    {
        typedef __attribute__((address_space(1))) int GI;
        typedef __attribute__((address_space(3))) int LI;
        for (int i = tid; i < PREDN; i += NTH) {
            GI* gp = (GI*)(uintptr_t)(predb + i);
            LI* lp = (LI*)(uint32_t)(uintptr_t)(&s_pred[i]);
            __builtin_amdgcn_global_load_async_to_lds_b32(gp, lp, 0, 0);
        }
    }
#else
    {
        const float2* __restrict__ p2 = (const float2*)predb;
        float2* s2 = (float2*)s_pred;
        for (int i = tid; i < PREDN / 2; i += NTH) s2[i] = p2[i];
    }
#endif

    // Zero the target grids while the DMA is in flight (disjoint LDS ranges).
    for (int i = tid; i < CELLS; i += NTH)         s_tobj[i] = 0.0f;
    for (int i = tid; i < CELLS * 4; i += NTH)     s_tbox[i] = 0.0f;
    for (int i = tid; i < CELLS * CCLS; i += NTH)  s_tcls[i] = 0.0f;

#if defined(ASYNC_LDS_B64) || defined(ASYNC_LDS_B32)
    asm volatile("s_wait_asynccnt 0" ::: "memory");
#endif
    __syncthreads();

    // Scatter targets (one lane per target; duplicates race like JAX scatter)
    if (tid < NTGT) {
        const float* tb = tboxes + ((size_t)b * NTGT + tid) * 4;
        float cx = tb[0], cy = tb[1], w = tb[2], h = tb[3];
        bool valid = (cx > 0.f) && (cx < 1.f) && (cy > 0.f) && (cy < 1.f) &&
                     (w > 0.f) && (h > 0.f);
        float wsv = valid ? w : 1.0f;
        float hsv = valid ? h : 1.0f;
        int gx = (int)(cx * (float)WDIM);
        gx = gx < 0 ? 0 : (gx > WDIM - 1 ? WDIM - 1 : gx);
        int gy = (int)(cy * (float)HDIM);
        gy = gy < 0 ? 0 : (gy > HDIM - 1 ? HDIM - 1 : gy);

        int   best = 0;
        float best_iou = -1.0f, aw_b = 1.0f, ah_b = 1.0f;
        for (int a = 0; a < AANC; ++a) {
            float aw = anchors[a * 2 + 0], ah = anchors[a * 2 + 1];
            float inter = fminf(wsv, aw) * fminf(hsv, ah);
            float iou = inter / (wsv * hsv + aw * ah - inter);
            if (iou > best_iou) { best_iou = iou; best = a; aw_b = aw; ah_b = ah; }
        }
        if (valid) {
            int cell = best * HW + gy * WDIM + gx;
            float tx = cx * (float)WDIM - (float)gx;
            float ty = cy * (float)HDIM - (float)gy;
            float tw = __logf(wsv / aw_b + 1e-16f);
            float th = __logf(hsv / ah_b + 1e-16f);
            s_tobj[cell] = 1.0f;
            s_tbox[cell * 4 + 0] = tx;
            s_tbox[cell * 4 + 1] = ty;
            s_tbox[cell * 4 + 2] = tw;
            s_tbox[cell * 4 + 3] = th;
            int lab = tlabels[(size_t)b * NTGT + tid];
            if (lab >= 0 && lab < CCLS) s_tcls[cell * CCLS + lab] = 1.0f;
        }
    }
    __syncthreads();

    // Per-cell loss accumulation (507 cells over 256 threads)
    float sum_all = 0.f, sum_pos = 0.f, cnt = 0.f, bb = 0.f, cl = 0.f;
    for (int idx = tid; idx < CELLS; idx += NTH) {
        int a   = idx / HW;
        int rem = idx - a * HW;
        const float* pc = &s_pred[a * CH * HW + rem];
        float to = s_tobj[idx];
        float l  = bce_logits(pc[0], to);
        sum_all += l;
        if (to > 0.f) {
            sum_pos += l;
            cnt     += 1.f;
#pragma unroll
            for (int i = 0; i < 4; ++i) {
                float d = pc[(1 + i) * HW] - s_tbox[idx * 4 + i];
                bb += d * d;
            }
#pragma unroll
            for (int c = 0; c < CCLS; ++c)
                cl += bce_logits(pc[(5 + c) * HW], s_tcls[idx * CCLS + c]);
        }
    }

    // wave32 butterfly reduction of 5 partials
#pragma unroll
    for (int m = 16; m >= 1; m >>= 1) {
        sum_all += __shfl_xor(sum_all, m, 32);
        sum_pos += __shfl_xor(sum_pos, m, 32);
        cnt     += __shfl_xor(cnt, m, 32);
        bb      += __shfl_xor(bb, m, 32);
        cl      += __shfl_xor(cl, m, 32);
    }
    const int wave = tid >> 5, lane = tid & 31;
    if (lane == 0) {
        s_red[wave * 5 + 0] = sum_all;
        s_red[wave * 5 + 1] = sum_pos;
        s_red[wave * 5 + 2] = cnt;
        s_red[wave * 5 + 3] = bb;
        s_red[wave * 5 + 4] = cl;
    }
    __syncthreads();

    if (tid == 0) {
        float SA = 0.f, SP = 0.f, CN = 0.f, BB = 0.f, CL = 0.f;
        for (int w8 = 0; w8 < NTH / 32; ++w8) {
            SA += s_red[w8 * 5 + 0];
            SP += s_red[w8 * 5 + 1];
            CN += s_red[w8 * 5 + 2];
            BB += s_red[w8 * 5 + 3];
            CL += s_red[w8 * 5 + 4];
        }
        float npos = CN;
        float pw   = ((float)CELLS - npos) / (npos + 1e-16f);
        float obj_b  = (SA - SP + SP * pw) * (1.0f / (float)CELLS);
        float bbox_b = npos > 0.f ? BB / (4.0f * npos + 1e-30f) : 0.0f;
        float cls_b  = npos > 0.f ? CL / ((float)CCLS * npos + 1e-30f) : 0.0f;
        ws_obj[b]  = obj_b;
        ws_bbox[b] = bbox_b;
        ws_cls[b]  = cls_b;
    }
}

// ---------------------------------------------------------------------------
// Kernel 2: 4096 -> scalar reduction using V_WMMA_F32_16X16X4_F32 as an exact
// f32 64-way summing primitive.  D = A x ones + C accumulates row sums; every
// one of the 64 A slots (2 VGPRs x 32 lanes) is added exactly once per wmma,
// so summing the 16 D rows yields the exact f32 tree-sum of the inputs.
// Wave w (0..2) reduces one of {obj, bbox, cls}; wave 3 runs identically on
// the cls array and its result is discarded (keeps EXEC all-1s, no divergence).
// ---------------------------------------------------------------------------
__global__ __launch_bounds__(128) void sgs_finalize(
    const float* __restrict__ ws_obj,
    const float* __restrict__ ws_bbox,
    const float* __restrict__ ws_cls,
    float* __restrict__ out)
{
    __shared__ float wsum[4];
    const int tid  = threadIdx.x;
    const int wave = tid >> 5;
    const int lane = tid & 31;

    const float* src = (wave == 0) ? ws_obj : (wave == 1) ? ws_bbox : ws_cls;

    v2f ones; ones.x = 1.0f; ones.y = 1.0f;
    v8f acc = {};
    for (int base = 0; base < BSZ; base += 64) {
        v2f a;
        a.x = src[base + lane];
        a.y = src[base + 32 + lane];
        acc = __builtin_amdgcn_wmma_f32_16x16x4_f32(
            /*neg_a=*/false, a, /*neg_b=*/false, ones,
            /*c_mod=*/(short)0, acc, /*reuse_a=*/false, /*reuse_b=*/false);
    }
    // Rows 0..7 live in acc[0..7] lanes 0..15; rows 8..15 in lanes 16..31.
    float t = acc[0] + acc[1] + acc[2] + acc[3] + acc[4] + acc[5] + acc[6] + acc[7];
    float s = __shfl(t, 0, 32) + __shfl(t, 16, 32);

    if (lane == 0 && wave < 3) wsum[wave] = s;
    __syncthreads();

    if (tid == 0) {
        const float inv = 1.0f / (float)BSZ;
        float obj = wsum[0] * inv;
        float bb  = wsum[1] * inv;
        float cl  = wsum[2] * inv;
        out[0] = 2.0f * obj + 5.0f * bb + 2.0f * cl;
        out[1] = obj;
        out[2] = bb;
        out[3] = cl;
    }
}

// ---------------------------------------------------------------------------
extern "C" void kernel_launch(void* const* d_in, const int* in_sizes, int n_in,
                              void* d_out, int out_size, void* d_ws, size_t ws_size,
                              hipStream_t stream) {
    (void)in_sizes; (void)n_in; (void)out_size; (void)ws_size;
    const float* pred = (const float*)d_in[0];
    const float* tb   = (const float*)d_in[1];
    const int*   tl   = (const int*)d_in[2];
    const float* an   = (const float*)d_in[3];

    float* ws      = (float*)d_ws;          // needs 3*4096 floats = 48 KB
    float* ws_obj  = ws;
    float* ws_bbox = ws + BSZ;
    float* ws_cls  = ws + 2 * BSZ;

    sgs_loss_main<<<BSZ, NTH, 0, stream>>>(pred, tb, tl, an, ws_obj, ws_bbox, ws_cls);
    sgs_finalize<<<1, 128, 0, stream>>>(ws_obj, ws_bbox, ws_cls, (float*)d_out);
}